// VectorQuantizer_13142599925854
// MI455X (gfx1250) — compile-verified
//
#include <hip/hip_runtime.h>

// ---------------------------------------------------------------------------
// VQ-VAE vector quantizer for MI455X (gfx1250, wave32, WMMA).
//   latents  [32768, 256] fp32, codebook [4096, 256] fp32
//   out: quantized [32768,256] fp32  ++  vq_loss scalar fp32
// Core: dist argmin via fp32 WMMA (V_WMMA_F32_16X16X4_F32), A resident in
// VGPRs; codebook streamed into double-buffered LDS with
// GLOBAL_LOAD_ASYNC_TO_LDS_B128 (ASYNCcnt) overlapping the WMMA chains.
// ---------------------------------------------------------------------------

typedef float v2f __attribute__((ext_vector_type(2)));
typedef float v4f __attribute__((ext_vector_type(4)));
typedef float v8f __attribute__((ext_vector_type(8)));

#define DDIM       256     // vector dimension
#define CT         32      // codes per LDS buffer
#define LDS_STRIDE 260     // dwords per code row (260%64==4 -> conflict-free b64 reads)
#define ROWS_PER_WG 128    // 8 waves * 16 rows

// LDS byte address of a generic pointer to LDS (flat aperture: addr[31:0]).
static __device__ __forceinline__ unsigned lds_addr32(const void* p) {
    return (unsigned)(unsigned long long)p;
}

// Issue one async global->LDS 16B copy (ASYNCcnt-tracked, no VGPR data).
static __device__ __forceinline__ void async_copy_b128(unsigned lds_off,
                                                       const void* gptr) {
    asm volatile("global_load_async_to_lds_b128 %0, %1, off"
                 :: "v"(lds_off), "v"((unsigned long long)gptr)
                 : "memory");
}

static __device__ __forceinline__ void wait_asynccnt0() {
    asm volatile("s_wait_asynccnt 0x0" ::: "memory");
}

// Stage CT codebook rows (CT*DDIM floats) into an LDS buffer, padded layout.
static __device__ __forceinline__ void stage_tile_async(float* dst,
                                                        const float* src,
                                                        int tid) {
    const unsigned base = lds_addr32(dst);
    #pragma unroll
    for (int u = tid; u < CT * DDIM / 4; u += 256) {   // 16B units
        int code = u >> 6;                 // 64 units per 256-float row
        int kk   = (u & 63) << 2;          // dword offset within row
        unsigned loff = base + (unsigned)((code * LDS_STRIDE + kk) * 4);
        async_copy_b128(loff, src + (size_t)u * 4);
    }
}

// ---------------------------------------------------------------------------
// Kernel 1: cnorm[k] = sum_d codebook[k][d]^2   (one wave per code)
// ---------------------------------------------------------------------------
__global__ void vq_cnorm_kernel(const float* __restrict__ cb,
                                float* __restrict__ cnorm, int kcodes) {
    int gwave = (blockIdx.x * blockDim.x + threadIdx.x) >> 5;
    int lane  = threadIdx.x & 31;
    if (gwave >= kcodes) return;
    const float* row = cb + (size_t)gwave * DDIM;
    float s = 0.f;
    #pragma unroll
    for (int i = 0; i < DDIM / 32; ++i) {
        float v = row[lane + i * 32];
        s += v * v;
    }
    #pragma unroll
    for (int off = 16; off > 0; off >>= 1) s += __shfl_xor(s, off, 32);
    if (lane == 0) cnorm[gwave] = s;
}

// ---------------------------------------------------------------------------
// Kernel 2: main — per 16-row tile: argmin_k (cnorm[k] - 2 x.c_k) via WMMA,
// then gather codebook rows, write quantized, accumulate (q-x)^2 partials.
// Dynamic LDS: [ buf0 | buf1 | wsum(8) ]
// ---------------------------------------------------------------------------
__global__ __launch_bounds__(256, 1)
void vq_main_kernel(const float* __restrict__ x, const float* __restrict__ cb,
                    const float* __restrict__ cnorm, float* __restrict__ out,
                    float* __restrict__ partial, int kcodes) {
    extern __shared__ float smem[];
    float* buf0 = smem;
    float* buf1 = smem + CT * LDS_STRIDE;
    float* wsum = smem + 2 * CT * LDS_STRIDE;

    const int tid   = threadIdx.x;
    const int lane  = tid & 31;
    const int wave  = tid >> 5;
    const int khalf = lane >> 4;            // 0: K pair {0,1}, 1: K pair {2,3}
    const int lrow  = lane & 15;            // A: matrix row / B: code column
    const int rowbase = blockIdx.x * ROWS_PER_WG + wave * 16;

    // ---- prologue: async-stage tile 0 while we load A fragments -----------
    stage_tile_async(buf0, cb, tid);

    // -------- load this wave's A operand (16 rows x 256) as WMMA fragments --
    // A 16x4 f32 layout: lanes 0-15 row M=lane hold K={0,1}; lanes 16-31 K={2,3}.
    v2f A[DDIM / 4];
    {
        const v2f* ap = (const v2f*)(x + (size_t)(rowbase + lrow) * DDIM + 2 * khalf);
        #pragma unroll
        for (int j = 0; j < DDIM / 4; ++j) A[j] = ap[2 * j];   // global_load_b64
    }

    float minv[8];
    int   mini[8];
    #pragma unroll
    for (int r = 0; r < 8; ++r) { minv[r] = 3.402823e38f; mini[r] = 0; }

    wait_asynccnt0();
    __syncthreads();

    // ---------------- K loop, double-buffered async codebook stream --------
    for (int kt = 0; kt < kcodes; kt += CT) {
        float* cur = ((kt / CT) & 1) ? buf1 : buf0;
        float* nxt = ((kt / CT) & 1) ? buf0 : buf1;
        if (kt + CT < kcodes) {
            stage_tile_async(nxt, cb + (size_t)(kt + CT) * DDIM, tid);
            __builtin_prefetch(cb + (size_t)(kt + 2 * CT) * DDIM, 0, 1);
        }

        #pragma unroll
        for (int ct = 0; ct < CT; ct += 16) {
            const int codeloc = ct + lrow;
            const int code    = kt + codeloc;
            // B 4x16 f32: lanes 0-15 col N=lane hold K={0,1}; lanes 16-31 K={2,3}
            const float* brow = cur + codeloc * LDS_STRIDE + 2 * khalf;

            v8f acc = {0.f, 0.f, 0.f, 0.f, 0.f, 0.f, 0.f, 0.f};
            #pragma unroll
            for (int j = 0; j < DDIM / 4; ++j) {
                v2f b = *(const v2f*)(brow + 4 * j);           // ds_load_b64
                acc = __builtin_amdgcn_wmma_f32_16x16x4_f32(
                          false, A[j], false, b, (short)0, acc, false, false);
            }
            const float cn = cnorm[code];
            // C layout: lane holds column `code`, rows r + 8*khalf, r=0..7
            #pragma unroll
            for (int r = 0; r < 8; ++r) {
                float dist = cn - 2.0f * acc[r];
                if (dist < minv[r]) { minv[r] = dist; mini[r] = code; } // first-index ties
            }
        }

        wait_asynccnt0();      // our slice of the next tile has landed
        __syncthreads();       // everyone's slices have landed / reads done
    }

    // ---- argmin reduction across the 16 lanes holding different columns ----
    #pragma unroll
    for (int r = 0; r < 8; ++r) {
        #pragma unroll
        for (int off = 1; off < 16; off <<= 1) {
            float ov = __shfl_xor(minv[r], off, 32);
            int   oi = __shfl_xor(mini[r], off, 32);
            if (ov < minv[r] || (ov == minv[r] && oi < mini[r])) {
                minv[r] = ov; mini[r] = oi;
            }
        }
    }
    // lanes 0-15 now hold argmin for rows r (in slot r), lanes 16-31 rows r+8.

    // ---------------- gather + write quantized, accumulate loss -------------
    float lsum = 0.f;
    #pragma unroll
    for (int m = 0; m < 16; ++m) {
        int bk  = __shfl(mini[m & 7], (m < 8) ? 0 : 16, 32);
        int row = rowbase + m;
        const v4f* crow = (const v4f*)(cb + (size_t)bk * DDIM);
        const v4f* xrow = (const v4f*)(x + (size_t)row * DDIM);
        v4f*       orow = (v4f*)(out + (size_t)row * DDIM);
        v4f c0 = crow[lane], c1 = crow[lane + 32];
        v4f x0 = xrow[lane], x1 = xrow[lane + 32];
        __builtin_nontemporal_store(c0, orow + lane);
        __builtin_nontemporal_store(c1, orow + lane + 32);
        v4f d0 = c0 - x0, d1 = c1 - x1;
        lsum += d0.x * d0.x + d0.y * d0.y + d0.z * d0.z + d0.w * d0.w;
        lsum += d1.x * d1.x + d1.y * d1.y + d1.z * d1.z + d1.w * d1.w;
    }
    #pragma unroll
    for (int off = 16; off > 0; off >>= 1) lsum += __shfl_xor(lsum, off, 32);
    if (lane == 0) wsum[wave] = lsum;
    __syncthreads();
    if (tid == 0) {
        float s = 0.f;
        #pragma unroll
        for (int w = 0; w < 8; ++w) s += wsum[w];   // fixed order: deterministic
        partial[blockIdx.x] = s;
    }
}

// ---------------------------------------------------------------------------
// Kernel 3: deterministic final loss reduction
// ---------------------------------------------------------------------------
__global__ void vq_finalize_kernel(const float* __restrict__ partial, int n,
                                   float* __restrict__ out_loss, float scale) {
    if (threadIdx.x == 0 && blockIdx.x == 0) {
        float s = 0.f;
        for (int i = 0; i < n; ++i) s += partial[i];
        *out_loss = s * scale;
    }
}

// ---------------------------------------------------------------------------
extern "C" void kernel_launch(void* const* d_in, const int* in_sizes, int n_in,
                              void* d_out, int out_size, void* d_ws, size_t ws_size,
                              hipStream_t stream) {
    const float* latents  = (const float*)d_in[0];
    const float* codebook = (const float*)d_in[1];
    float* out = (float*)d_out;

    const int rows   = in_sizes[0] / DDIM;   // 32768
    const int kcodes = in_sizes[1] / DDIM;   // 4096
    const int nwg    = rows / ROWS_PER_WG;   // 256

    float* cnorm   = (float*)d_ws;           // kcodes floats
    float* partial = cnorm + kcodes;         // nwg floats

    // 1) code norms (one wave per code)
    {
        int blocks = (kcodes * 32 + 255) / 256;
        vq_cnorm_kernel<<<blocks, 256, 0, stream>>>(codebook, cnorm, kcodes);
    }
    // 2) main WMMA distance/argmin + gather + partial loss
    size_t shbytes = (size_t)(2 * CT * LDS_STRIDE + 8) * sizeof(float); // 66,592 B
    vq_main_kernel<<<nwg, 256, shbytes, stream>>>(latents, codebook, cnorm, out,
                                                  partial, kcodes);
    // 3) loss = 1.25 * mean((q - x)^2)
    float scale = (float)(1.25 / ((double)rows * (double)DDIM));
    vq_finalize_kernel<<<1, 32, 0, stream>>>(partial, nwg,
                                             out + (out_size - 1), scale);
}